// SO3NLLLoss_49624052138459
// MI455X (gfx1250) — compile-verified
//
#include <hip/hip_runtime.h>
#include <hip/hip_bf16.h>
#include <cstdint>

// SO(3) NLL loss: nll[b] = 0.5 * r^T Rinv r - 0.5*log(det(Rinv)),
// r = so3_log(C_est C_target^T).
//
// Roofline: 112 B/elem vs ~1.3 FLOP/B -> HBM-bound (~9.6us floor at 23.3 TB/s
// for B=2M). Per-batch 3x3 matrices map to per-lane scalar math under wave32;
// a 16x16 WMMA tile would be <4% utilized and add shuffle traffic, so the
// CDNA5 path that pays here is data movement:
//   - async global->LDS b128 staging (ASYNCcnt, bypasses VGPRs)
//   - conflict-free stride-9 LDS reads (gcd(9,64 banks)=1)
//   - NT store of the streamed-once output.

#define SO3_EPS 1e-7f

__device__ __forceinline__ void async_ld_b128(uint32_t lds_off, const void* gaddr) {
    // VDST = LDS byte offset, VADDR = 64-bit global address (VGPR pair).
    asm volatile("global_load_async_to_lds_b128 %0, %1, off"
                 :: "v"(lds_off), "v"(gaddr) : "memory");
}
__device__ __forceinline__ void async_ld_b32(uint32_t lds_off, const void* gaddr) {
    asm volatile("global_load_async_to_lds_b32 %0, %1, off"
                 :: "v"(lds_off), "v"(gaddr) : "memory");
}
__device__ __forceinline__ void wait_async_zero() {
    asm volatile("s_wait_asynccnt 0" ::: "memory");
}

__global__ __launch_bounds__(256) void so3_nll_kernel(
    const float* __restrict__ Cest,
    const float* __restrict__ Ctgt,
    const float* __restrict__ Rinv,
    float* __restrict__ out,
    int B)
{
    // 256 elements/block * 9 floats * 3 arrays = 27648 B of LDS (of 320KB/WGP).
    __shared__ alignas(16) float lds[3 * 2304];

    const int t         = (int)threadIdx.x;
    const int blockBase = (int)blockIdx.x * 256;
    int rem = B - blockBase;
    if (rem > 256) rem = 256;

    const uint32_t ldsBase = (uint32_t)(uintptr_t)(&lds[0]);
    const size_t   gOff    = (size_t)blockBase * 9;
    const float* srcs[3] = { Cest + gOff, Ctgt + gOff, Rinv + gOff };

    if (rem == 256) {
        // Fast path (all but the final block): 576 float4 per array,
        // fully unrolled: i = t, t+256, and lanes t<64 take i = t+512.
#pragma unroll
        for (int a = 0; a < 3; ++a) {
            const float*   s  = srcs[a];
            const uint32_t lb = ldsBase + (uint32_t)a * 9216u;
            async_ld_b128(lb + (uint32_t)t * 16u,          (const void*)(s + (size_t)t * 4));
            async_ld_b128(lb + (uint32_t)(t + 256) * 16u,  (const void*)(s + (size_t)(t + 256) * 4));
        }
        if (t < 64) {
#pragma unroll
            for (int a = 0; a < 3; ++a) {
                const float*   s  = srcs[a];
                const uint32_t lb = ldsBase + (uint32_t)a * 9216u;
                async_ld_b128(lb + (uint32_t)(t + 512) * 16u, (const void*)(s + (size_t)(t + 512) * 4));
            }
        }
    } else {
        // Tail block (at most one per grid): generic guarded staging.
        const int nFloats = rem * 9;
        const int nF4     = nFloats >> 2;
        const int nTail   = nFloats & 3;
#pragma unroll
        for (int a = 0; a < 3; ++a) {
            const float*   s  = srcs[a];
            const uint32_t lb = ldsBase + (uint32_t)a * 9216u;
            for (int i = t; i < nF4; i += 256)
                async_ld_b128(lb + (uint32_t)i * 16u, (const void*)(s + (size_t)i * 4));
            for (int i = t; i < nTail; i += 256)
                async_ld_b32(lb + (uint32_t)(nF4 * 4 + i) * 4u, (const void*)(s + nF4 * 4 + i));
        }
    }

    wait_async_zero();   // drain this wave's ASYNCcnt
    __syncthreads();     // all waves drained -> staged data visible

    if (t < rem) {
        // Stride-9 word access across lanes: conflict-free on 64 banks.
        const float* E = &lds[t * 9];            // C_est rows
        const float* T = &lds[2304 + t * 9];     // C_target rows
        const float* R = &lds[4608 + t * 9];     // Rinv rows

        // dC = E * T^T ; need trace + 6 off-diagonals (row_i(E) . row_j(T))
        float tr  = E[0]*T[0] + E[1]*T[1] + E[2]*T[2]
                  + E[3]*T[3] + E[4]*T[4] + E[5]*T[5]
                  + E[6]*T[6] + E[7]*T[7] + E[8]*T[8];
        float d21 = E[6]*T[3] + E[7]*T[4] + E[8]*T[5];
        float d12 = E[3]*T[6] + E[4]*T[7] + E[5]*T[8];
        float d02 = E[0]*T[6] + E[1]*T[7] + E[2]*T[8];
        float d20 = E[6]*T[0] + E[7]*T[1] + E[8]*T[2];
        float d10 = E[3]*T[0] + E[4]*T[1] + E[5]*T[2];
        float d01 = E[0]*T[3] + E[1]*T[4] + E[2]*T[5];

        float v0 = d21 - d12;
        float v1 = d02 - d20;
        float v2 = d10 - d01;

        // so3_log coefficient.  After the clip, angle >= ~4.47e-4 > 1e-6,
        // so the reference's small-angle Taylor branch is unreachable.
        float c = 0.5f * (tr - 1.0f);
        c = fminf(fmaxf(c, -1.0f + SO3_EPS), 1.0f - SO3_EPS);
        float angle = acosf(c);
        float coef  = 0.5f * angle / sinf(angle);

        float r0 = coef * v0;
        float r1 = coef * v1;
        float r2 = coef * v2;

        // 0.5 * r^T Rinv r
        float q = r0 * (R[0]*r0 + R[1]*r1 + R[2]*r2)
                + r1 * (R[3]*r0 + R[4]*r1 + R[5]*r2)
                + r2 * (R[6]*r0 + R[7]*r1 + R[8]*r2);

        // logdet(Rinv) via closed-form 3x3 determinant (SPD by construction)
        float det = R[0] * (R[4]*R[8] - R[5]*R[7])
                  - R[1] * (R[3]*R[8] - R[5]*R[6])
                  + R[2] * (R[3]*R[7] - R[4]*R[6]);

        float nll = 0.5f * q - 0.5f * logf(det);

        // Result is streamed once, never re-read: NT store (TH=NT).
        __builtin_nontemporal_store(nll, out + blockBase + t);
    }
}

extern "C" void kernel_launch(void* const* d_in, const int* in_sizes, int n_in,
                              void* d_out, int out_size, void* d_ws, size_t ws_size,
                              hipStream_t stream) {
    const float* Cest = (const float*)d_in[0];   // [B,3,3] f32
    const float* Ctgt = (const float*)d_in[1];   // [B,3,3] f32
    const float* Rinv = (const float*)d_in[2];   // [B,3,3] f32
    float* out = (float*)d_out;                  // [B] f32

    const int B = in_sizes[0] / 9;
    const int blocks = (B + 255) / 256;
    so3_nll_kernel<<<blocks, 256, 0, stream>>>(Cest, Ctgt, Rinv, out, B);
}